// GINConvNet_46677704573590
// MI455X (gfx1250) — compile-verified
//
#include <hip/hip_runtime.h>

#define NN 200000
#define NE 2000000
#define NG 4096

typedef float f32x2 __attribute__((ext_vector_type(2)));
typedef float f32x8 __attribute__((ext_vector_type(8)));

// ---------------- elementwise / scatter kernels ----------------

__global__ void k_zero(float* __restrict__ p, long long n) {
  long long i = (long long)blockIdx.x * blockDim.x + threadIdx.x;
  if (i < n) p[i] = 0.0f;
}

// h[n][k] = (k < Kx) ? x[n][k] : 0   (row strides ldx -> ldh); scalar (layer 0)
__global__ void k_init_h(const float* __restrict__ x, int ldx, int Kx,
                         float* __restrict__ h, int ldh, long long total) {
  long long i = (long long)blockIdx.x * blockDim.x + threadIdx.x;
  if (i >= total) return;
  int k = (int)(i % ldh);
  long long n = i / ldh;
  h[i] = (k < Kx) ? x[n * (long long)ldx + k] : 0.0f;
}

// vectorized copy (16-B aligned buffers), total in float4 units
__global__ void k_copy4(const float4* __restrict__ src, float4* __restrict__ dst,
                        long long total4) {
  long long i = (long long)blockIdx.x * blockDim.x + threadIdx.x;
  if (i < total4) dst[i] = src[i];
}

// dst (rd x cols) = src (rs x cols) zero-padded in rows; total = rd*cols
__global__ void k_pad_rows(const float* __restrict__ src, int rs, int cols,
                           float* __restrict__ dst, long long total) {
  long long i = (long long)blockIdx.x * blockDim.x + threadIdx.x;
  if (i >= total) return;
  int r = (int)(i / cols);
  dst[i] = (r < rs) ? src[i] : 0.0f;
}

// scalar edge scatter (layer 0: 78-wide rows, not 16-B aligned)
__global__ void k_edge_scatter(const float* __restrict__ x, int ldx,
                               const int* __restrict__ srcv,
                               const int* __restrict__ dstv,
                               float* __restrict__ h, int ldh, int K,
                               long long total) {
  long long i = (long long)blockIdx.x * blockDim.x + threadIdx.x;
  if (i >= total) return;
  int k = (int)(i % K);
  long long e = i / K;
  int s = srcv[e];
  int d = dstv[e];
  atomicAdd(&h[(long long)d * ldh + k], x[(long long)s * ldx + k]);
}

// vectorized edge scatter for 32-wide rows: thread = (edge, quad)
__global__ void k_edge_scatter4(const float* __restrict__ x,
                                const int* __restrict__ srcv,
                                const int* __restrict__ dstv,
                                float* __restrict__ h, long long total) {
  long long i = (long long)blockIdx.x * blockDim.x + threadIdx.x;
  if (i >= total) return;
  int q = (int)(i & 7);        // quad index within 32 cols
  long long e = i >> 3;
  int s = srcv[e];
  int d = dstv[e];
  float4 v = *(const float4*)(x + (long long)s * 32 + q * 4);
  float* hp = h + (long long)d * 32 + q * 4;
  atomicAdd(hp + 0, v.x);
  atomicAdd(hp + 1, v.y);
  atomicAdd(hp + 2, v.z);
  atomicAdd(hp + 3, v.w);
}

// scale/shift from accumulated (sum, sumsq) column stats
__global__ void k_bn_finalize(const float* __restrict__ stats,
                              const float* __restrict__ gamma,
                              const float* __restrict__ beta,
                              float* __restrict__ ss, float invN) {
  int k = threadIdx.x;  // 32 threads
  float mean = stats[k] * invN;
  float var  = stats[32 + k] * invN - mean * mean;
  float sc   = gamma[k] * rsqrtf(var + 1e-5f);
  ss[k]      = sc;
  ss[32 + k] = beta[k] - mean * sc;
}

// x = y*scale + shift, vectorized; total4 = N*32/4
__global__ void k_bn_apply4(const float4* __restrict__ y,
                            const float* __restrict__ ss,
                            float4* __restrict__ xo, long long total4) {
  long long i = (long long)blockIdx.x * blockDim.x + threadIdx.x;
  if (i >= total4) return;
  int k4 = (int)(i & 7) * 4;
  float4 v = y[i];
  float4 o;
  o.x = v.x * ss[k4 + 0] + ss[32 + k4 + 0];
  o.y = v.y * ss[k4 + 1] + ss[32 + k4 + 1];
  o.z = v.z * ss[k4 + 2] + ss[32 + k4 + 2];
  o.w = v.w * ss[k4 + 3] + ss[32 + k4 + 3];
  xo[i] = o;
}

// pooled[batch[n]][k4..k4+3] += x[n][k4..k4+3]
__global__ void k_pool4(const float* __restrict__ xv, const int* __restrict__ batch,
                        float* __restrict__ pooled, long long total) {
  long long i = (long long)blockIdx.x * blockDim.x + threadIdx.x;
  if (i >= total) return;
  int q = (int)(i & 7);
  long long n = i >> 3;
  float4 v = *(const float4*)(xv + n * 32 + q * 4);
  float* pp = pooled + (long long)batch[n] * 32 + q * 4;
  atomicAdd(pp + 0, v.x);
  atomicAdd(pp + 1, v.y);
  atomicAdd(pp + 2, v.z);
  atomicAdd(pp + 3, v.w);
}

// out[r] = y2[r,:32] . Wo + bo   (N=1 column doesn't fit a WMMA tile)
__global__ void k_out_head(const float* __restrict__ y, const float* __restrict__ Wo,
                           const float* __restrict__ bo, float* __restrict__ out, int M) {
  int r = blockIdx.x * blockDim.x + threadIdx.x;
  if (r >= M) return;
  float s = bo[0];
#pragma unroll
  for (int k = 0; k < 32; ++k) s += y[(long long)r * 32 + k] * Wo[k];
  out[r] = s;
}

// ---------------- generic f32 WMMA GEMM with async-LDS A staging ----------------
// C[M x N] = A[M x K] (lda) @ B[K x N] (ldb) + bias[N], optional ReLU.
// Each wave owns one 16x16 C tile (V_WMMA_F32_16X16X4_F32).
// The 16xK A-panel is staged in K-chunks of 64 into a private LDS region via
// global_load_async_to_lds_b128 (coalesced 16-B transfers, ASYNCcnt), then A
// fragments are read from LDS with a 68-float row pitch (16-B aligned rows,
// conflict-free banks). B fragments load directly (contiguous across lanes).
// The kk-loop is unrolled 4x so loads batch into clauses and multiple WMMAs
// retire against staggered waits instead of one full stall per WMMA.
// If stats != nullptr (N==32 only) also accumulates column sum/sumsq via an
// LDS reduction. Requires M%16==0, N%16==0, K%4==0, lda%4==0, 16-B aligned A.
#define KC 64
#define APITCH 68  // floats; 272 B row pitch: 16-B aligned, bank-conflict free

__global__ void k_gemm_wmma(const float* __restrict__ A, int lda,
                            const float* __restrict__ B, int ldb,
                            const float* __restrict__ bias,
                            float* __restrict__ C, int ldc,
                            int M, int N, int K, int relu,
                            float* __restrict__ stats) {
  __shared__ __align__(16) float s_a[4][16 * APITCH];
  __shared__ float s_acc[64];
  const int lane   = threadIdx.x;          // 0..31 (wave32)
  const int wid    = blockIdx.x * blockDim.y + threadIdx.y;
  const int ntiles = N >> 4;
  const int tiles  = (M >> 4) * ntiles;
  const bool active = (wid < tiles);       // wave-uniform
  const int half = lane >> 4;              // 0: lanes 0-15, 1: lanes 16-31
  const int l    = lane & 15;

  if (stats) {
    int t = threadIdx.y * 32 + lane;
    if (t < 64) s_acc[t] = 0.0f;
    __syncthreads();
  }

  if (active) {
    const int m0  = (wid / ntiles) << 4;
    const int n0  = (wid % ntiles) << 4;
    const int col = n0 + l;

    float* sw = &s_a[threadIdx.y][0];
    const unsigned sa_base = (unsigned)(size_t)sw;  // LDS byte offset of this wave's panel
    // B fragment (4x16): VGPR0 = K0 | K2, VGPR1 = K1 | K3, N = lane%16
    const float* bp = B + (long long)(half * 2) * ldb + col;
    const int abase = l * APITCH + half * 2;        // A fragment base within LDS panel

    f32x8 c = {};
    for (int k0 = 0; k0 < K; k0 += KC) {
      const int Kc = (K - k0 < KC) ? (K - k0) : KC;
      const int nq = Kc >> 2;                       // 16-B quads per row
      const int tq = nq << 4;                       // quads in 16-row panel
      // stage A[m0:m0+16, k0:k0+Kc] -> LDS, coalesced async b128 transfers
      for (int f = lane; f < tq; f += 32) {
        const int r = f / nq;
        const int q = f - r * nq;
        const unsigned lds = sa_base + (unsigned)(r * (APITCH * 4) + q * 16);
        const float* g = A + (long long)(m0 + r) * lda + k0 + q * 4;
        asm volatile("global_load_async_to_lds_b128 %0, %1, off"
                     :: "v"(lds), "v"(g) : "memory");
      }
      asm volatile("s_wait_asynccnt 0x0" ::: "memory");

#pragma unroll 4
      for (int kk = 0; kk < Kc; kk += 4) {
        f32x2 av, bv;
        // A fragment (16x4): VGPR0 = K0 | K2, VGPR1 = K1 | K3, M = lane%16
        av.x = sw[abase + kk];
        av.y = sw[abase + kk + 1];
        const long long krow = (long long)(k0 + kk) * ldb;
        bv.x = bp[krow];
        bv.y = bp[krow + ldb];
        c = __builtin_amdgcn_wmma_f32_16x16x4_f32(
            /*neg_a=*/false, av, /*neg_b=*/false, bv,
            /*c_mod=*/(short)0, c, /*reuse_a=*/false, /*reuse_b=*/false);
      }
    }

    const float bb = bias ? bias[col] : 0.0f;
#pragma unroll
    for (int r = 0; r < 8; ++r) {
      // C/D layout: VGPR r -> row m0 + r (lanes 0-15) / m0 + 8 + r (lanes 16-31)
      const int row = m0 + r + half * 8;
      float v = c[r] + bb;
      if (relu) v = fmaxf(v, 0.0f);
      C[(long long)row * ldc + col] = v;
      if (stats && col < 32) {
        atomicAdd(&s_acc[col], v);
        atomicAdd(&s_acc[32 + col], v * v);
      }
    }
  }

  if (stats) {
    __syncthreads();
    int t = threadIdx.y * 32 + lane;
    if (t < 64) atomicAdd(&stats[t], s_acc[t]);
  }
}

// ---------------- host orchestration ----------------

static inline unsigned g1d(long long total) {
  return (unsigned)((total + 255) / 256);
}

extern "C" void kernel_launch(void* const* d_in, const int* in_sizes, int n_in,
                              void* d_out, int out_size, void* d_ws, size_t ws_size,
                              hipStream_t stream) {
  (void)in_sizes; (void)n_in; (void)out_size; (void)ws_size;

  // setup_inputs() dict order, pytrees flattened in insertion order:
  // 0 x1, 1 ei1, 2 b1, 3 x2, 4 ei2, 5 b2, 6 target,
  // params1: mlp (W1,b1,W2,b2)x5 -> 7..26, bn (g,b)x5 -> 27..36, fc -> 37,38
  // params2: 39..70 (same layout), head: fc_xt 71,72  fc1 73,74  fc2 75,76  out 77,78
  const float* xin[2]   = {(const float*)d_in[0], (const float*)d_in[3]};
  const int*   ei[2]    = {(const int*)d_in[1], (const int*)d_in[4]};
  const int*   batch[2] = {(const int*)d_in[2], (const int*)d_in[5]};
  const float* target   = (const float*)d_in[6];
  const int pbase[2]    = {7, 39};

  // workspace layout (floats; every offset is a multiple of 32 floats -> 128-B aligned)
  float* ws    = (float*)d_ws;
  float* hbuf  = ws;  ws += (long long)NN * 80;   // aggregated features (padded)
  float* mid   = ws;  ws += (long long)NN * 32;   // relu(H@W1+b1)
  float* ybuf  = ws;  ws += (long long)NN * 32;   // relu(mid@W2+b2)
  float* xbuf  = ws;  ws += (long long)NN * 32;   // batch-normed features
  float* w1pad = ws;  ws += 80 * 32;              // zero-padded layer-0 W1
  float* stats = ws;  ws += 64;                   // col sum / sumsq
  float* ssbuf = ws;  ws += 64;                   // BN scale / shift
  float* pooled= ws;  ws += (long long)NG * 32;
  float* xc    = ws;  ws += (long long)NG * 256;  // [h1 | h2 | xt]
  float* y1h   = ws;  ws += (long long)NG * 128;
  float* y2h   = ws;  ws += (long long)NG * 32;

  auto gemm = [&](const float* A, int lda, const float* B, int ldb,
                  const float* bias, float* C, int ldc,
                  int M, int N, int K, int relu, float* st) {
    int tiles = (M / 16) * (N / 16);
    dim3 grid((unsigned)((tiles + 3) / 4));
    dim3 block(32, 4);
    hipLaunchKernelGGL(k_gemm_wmma, grid, block, 0, stream,
                       A, lda, B, ldb, bias, C, ldc, M, N, K, relu, st);
  };

  for (int b = 0; b < 2; ++b) {
    const int pb    = pbase[b];
    const int* srcv = ei[b];
    const int* dstv = ei[b] + NE;

    for (int l = 0; l < 5; ++l) {
      const float* W1  = (const float*)d_in[pb + 4 * l + 0];
      const float* b1v = (const float*)d_in[pb + 4 * l + 1];
      const float* W2  = (const float*)d_in[pb + 4 * l + 2];
      const float* b2v = (const float*)d_in[pb + 4 * l + 3];
      const float* gma = (const float*)d_in[pb + 20 + 2 * l];
      const float* bta = (const float*)d_in[pb + 20 + 2 * l + 1];

      if (l == 0) {
        // pad layer-0 W1 (78x32 -> 80x32)
        const long long tp = 80 * 32;
        hipLaunchKernelGGL(k_pad_rows, dim3(g1d(tp)), dim3(256), 0, stream,
                           W1, 78, 32, w1pad, tp);
        // h = [x | 0], scalar (78-wide rows not 16-B aligned)
        const long long th = (long long)NN * 80;
        hipLaunchKernelGGL(k_init_h, dim3(g1d(th)), dim3(256), 0, stream,
                           xin[b], 78, 78, hbuf, 80, th);
        const long long te = (long long)NE * 78;
        hipLaunchKernelGGL(k_edge_scatter, dim3(g1d(te)), dim3(256), 0, stream,
                           xin[b], 78, srcv, dstv, hbuf, 80, 78, te);
        gemm(hbuf, 80, w1pad, 32, b1v, mid, 32, NN, 32, 80, 1, nullptr);
      } else {
        // h = x (vector copy), then vectorized scatter
        const long long t4 = (long long)NN * 8;
        hipLaunchKernelGGL(k_copy4, dim3(g1d(t4)), dim3(256), 0, stream,
                           (const float4*)xbuf, (float4*)hbuf, t4);
        const long long te4 = (long long)NE * 8;
        hipLaunchKernelGGL(k_edge_scatter4, dim3(g1d(te4)), dim3(256), 0, stream,
                           xbuf, srcv, dstv, hbuf, te4);
        gemm(hbuf, 32, W1, 32, b1v, mid, 32, NN, 32, 32, 1, nullptr);
      }

      // y = relu(mid @ W2 + b2), fused BN stats
      hipLaunchKernelGGL(k_zero, dim3(1), dim3(64), 0, stream, stats, (long long)64);
      gemm(mid, 32, W2, 32, b2v, ybuf, 32, NN, 32, 32, 1, stats);
      // batchnorm
      hipLaunchKernelGGL(k_bn_finalize, dim3(1), dim3(32), 0, stream,
                         stats, gma, bta, ssbuf, 1.0f / (float)NN);
      const long long tn4 = (long long)NN * 8;
      hipLaunchKernelGGL(k_bn_apply4, dim3(g1d(tn4)), dim3(256), 0, stream,
                         (const float4*)ybuf, ssbuf, (float4*)xbuf, tn4);
    }

    // segment-sum pooling over graphs
    const long long tpd = (long long)NG * 32;
    hipLaunchKernelGGL(k_zero, dim3(g1d(tpd)), dim3(256), 0, stream, pooled, tpd);
    const long long tn4 = (long long)NN * 8;
    hipLaunchKernelGGL(k_pool4, dim3(g1d(tn4)), dim3(256), 0, stream,
                       xbuf, batch[b], pooled, tn4);

    // branch FC 32->64 + ReLU, written into xc columns [b*64, b*64+64)
    const float* Wf = (const float*)d_in[pb + 30];
    const float* bf = (const float*)d_in[pb + 31];
    gemm(pooled, 32, Wf, 64, bf, xc + b * 64, 256, NG, 64, 32, 1, nullptr);
  }

  // xt = target @ Wt + bt (no activation), into xc columns [128, 256)
  gemm(target, 1000, (const float*)d_in[71], 128, (const float*)d_in[72],
       xc + 128, 256, NG, 128, 1000, 0, nullptr);
  // fc1: 256 -> 128, ReLU
  gemm(xc, 256, (const float*)d_in[73], 128, (const float*)d_in[74],
       y1h, 128, NG, 128, 256, 1, nullptr);
  // fc2: 128 -> 32, ReLU
  gemm(y1h, 128, (const float*)d_in[75], 32, (const float*)d_in[76],
       y2h, 32, NG, 32, 128, 1, nullptr);
  // out: 32 -> 1
  hipLaunchKernelGGL(k_out_head, dim3(g1d(NG)), dim3(256), 0, stream,
                     y2h, (const float*)d_in[77], (const float*)d_in[78],
                     (float*)d_out, NG);
}